// HyperSpectralIntegrator_83210696393012
// MI455X (gfx1250) — compile-verified
//
#include <hip/hip_runtime.h>

// gfx1250 is wave32-only; WMMA f32 16x16x4 keeps the reduction in full fp32.
typedef float v2f __attribute__((ext_vector_type(2)));
typedef float v8f __attribute__((ext_vector_type(8)));

#define NSMPL  4096
#define NBANDS 10

__global__ __launch_bounds__(128) void trapz_mix_wmma(
    const float* __restrict__ spectra,      // [bsz, NSMPL]
    const float* __restrict__ trans,        // [bsz, NBANDS, NSMPL]
    const float* __restrict__ nsmpl_band,   // [NBANDS]
    float* __restrict__ out)                // [bsz, NBANDS]
{
    const int b    = blockIdx.x;
    const int tid  = threadIdx.x;
    const int lane = tid & 31;        // wave32
    const int wave = tid >> 5;        // 4 waves per block

    // WMMA f32 16x16x4 A-matrix layout (ISA 7.12.2):
    //   lanes 0-15 : M = lane,    VGPR0 = K0, VGPR1 = K1
    //   lanes 16-31: M = lane-16, VGPR0 = K2, VGPR1 = K3
    // -> each lane loads 2 consecutive floats of its band row at n0 + 2*half.
    // B (4x16) follows the same half-wave striping ({K0|K2},{K1|K3}); the
    // identical 8B load of spectra broadcasts s[n0+k] across all 16 columns.
    const int row  = lane & 15;
    const int rowc = row < NBANDS ? row : (NBANDS - 1);  // pad rows read valid mem
    const int half = lane >> 4;

    const float* tb   = trans   + (size_t)b * NBANDS * NSMPL;
    const float* arow = tb      + (size_t)rowc * NSMPL + 2 * half;
    const float* srow = spectra + (size_t)b * NSMPL    + 2 * half;

    const int nbase = wave * (NSMPL / 4);   // 1024 samples per wave

    v8f c0 = {};
    v8f c1 = {};
    for (int i = 0; i < NSMPL / 4; i += 8) {
        if ((i & 63) == 0) {
            // stream-ahead prefetch of this lane's band row (256B line)
            __builtin_prefetch(arow + nbase + i + 256, 0, 0);
        }
        // trans: 640MB single-use stream -> non-temporal
        v2f a0 = __builtin_nontemporal_load((const v2f*)(arow + nbase + i));
        v2f a1 = __builtin_nontemporal_load((const v2f*)(arow + nbase + i + 4));
        // spectra: reused across both lane halves / bands -> regular temporal
        v2f b0 = *(const v2f*)(srow + nbase + i);
        v2f b1 = *(const v2f*)(srow + nbase + i + 4);
        // (neg_a, A, neg_b, B, c_mod, C, reuse_a, reuse_b)
        c0 = __builtin_amdgcn_wmma_f32_16x16x4_f32(false, a0, false, b0,
                                                   (short)0, c0, false, false);
        c1 = __builtin_amdgcn_wmma_f32_16x16x4_f32(false, a1, false, b1,
                                                   (short)0, c1, false, false);
    }
    v8f c = c0 + c1;

    // D layout: VGPR v -> M=v (lanes 0-15, N=lane) / M=v+8 (lanes 16-31).
    // All N columns are identical by construction; take column 0 of each half.
    __shared__ float red[4 * 16];
    if (lane == 0) {
#pragma unroll
        for (int m = 0; m < 8; ++m) red[wave * 16 + m] = c[m];
    } else if (lane == 16) {
        red[wave * 16 + 8] = c[0];
        red[wave * 16 + 9] = c[1];
    }
    __syncthreads();

    if (tid < NBANDS) {
        float s = red[tid] + red[16 + tid] + red[32 + tid] + red[48 + tid];
        // trapezoid endpoint correction: full-weight sum minus 0.5*(y0 + yN-1)
        const float* trow = tb + (size_t)tid * NSMPL;
        const float* sp   = spectra + (size_t)b * NSMPL;
        float corr = 0.5f * (sp[0] * trow[0] + sp[NSMPL - 1] * trow[NSMPL - 1]);
        out[(size_t)b * NBANDS + tid] = (s - corr) / nsmpl_band[tid];
    }
}

extern "C" void kernel_launch(void* const* d_in, const int* in_sizes, int n_in,
                              void* d_out, int out_size, void* d_ws, size_t ws_size,
                              hipStream_t stream) {
    (void)n_in; (void)out_size; (void)d_ws; (void)ws_size;
    const float* spectra = (const float*)d_in[0];
    const float* trans   = (const float*)d_in[1];
    const float* nsmpl   = (const float*)d_in[2];
    float* out           = (float*)d_out;

    const int bsz = in_sizes[0] / NSMPL;   // 4096
    trapz_mix_wmma<<<bsz, 128, 0, stream>>>(spectra, trans, nsmpl, out);
}